// ConvTemporalGraphical_13898514170199
// MI455X (gfx1250) — compile-verified
//
#include <hip/hip_runtime.h>
#include <math.h>

typedef float v2f __attribute__((ext_vector_type(2)));
typedef float v4f __attribute__((ext_vector_type(4)));
typedef float v8f __attribute__((ext_vector_type(8)));

#define LDSS 65  // padded row stride (floats) to break bank conflicts on column access

__global__ __launch_bounds__(256)
void stgcn_fused_kernel(const float* __restrict__ x,
                        const float* __restrict__ A,
                        const float* __restrict__ W,
                        const float* __restrict__ bias,
                        float* __restrict__ outX,
                        float* __restrict__ outA)
{
    // LDS buffers (aliased across phases):
    __shared__ float smem0[64 * LDSS];   // phase1: W      | phase2+: S
    __shared__ float smem1[64 * LDSS];   // phase1: X      | phase2 : A_k tile
    __shared__ float smemXp[64 * LDSS];  // Xp = W @ X + b
    __shared__ float lBias[64];
    __shared__ float lRed[256];
    __shared__ float lDinv[64];

    float* lW  = smem0;  float* lS = smem0;
    float* lX  = smem1;  float* lA = smem1;
    float* lXp = smemXp;

    const int tid = threadIdx.x;         // 0..255
    const int bt  = blockIdx.x;          // 0..2047
    const int b   = bt >> 5;             // B = 64
    const int t   = bt & 31;             // T = 32

    // ---- Stage W (64x64, contiguous) and bias into LDS --------------------
    {
        const v4f* gW = (const v4f*)W;
        #pragma unroll
        for (int i = 0; i < 4; ++i) {
            int idx4 = tid + i * 256;            // 1024 float4 total
            v4f w = gW[idx4];
            int e = idx4 << 2;
            int o = e >> 6, c = e & 63;
            lW[o * LDSS + c + 0] = w.x;
            lW[o * LDSS + c + 1] = w.y;
            lW[o * LDSS + c + 2] = w.z;
            lW[o * LDSS + c + 3] = w.w;
        }
        if (tid < 64) lBias[tid] = bias[tid];
    }
    // ---- Stage X tile: x[b, c, t, :] for c = 0..63 ------------------------
    {
        #pragma unroll
        for (int i = 0; i < 4; ++i) {
            int idx4 = tid + i * 256;
            int c   = idx4 >> 4;                 // 16 float4 per row
            int m4  = (idx4 & 15) << 2;
            const v4f* src = (const v4f*)(x + ((size_t)((b * 64 + c) * 32 + t) << 6));
            v4f v = src[idx4 & 15];
            lX[c * LDSS + m4 + 0] = v.x;
            lX[c * LDSS + m4 + 1] = v.y;
            lX[c * LDSS + m4 + 2] = v.z;
            lX[c * LDSS + m4 + 3] = v.w;
        }
    }
    __syncthreads();

    // ---- WMMA tiling setup (wave32) ---------------------------------------
    const int wave = tid >> 5, lane = tid & 31;
    const int half = lane >> 4, l16 = lane & 15;
    const int tr  = wave & 3;            // D tile row 0..3
    const int tcb = (wave >> 2) << 1;    // D tile col base: 0 or 2
    const int rowA = 16 * tr + l16;      // A-operand row (M)
    const int c0 = 16 * tcb + l16;       // B-operand col, tile 0
    const int c1 = c0 + 16;              // B-operand col, tile 1

    // ---- GEMM1: Xp = W @ X + bias  (16x16x4 f32 WMMA) ---------------------
    v8f acc0, acc1;
    #pragma unroll
    for (int v = 0; v < 8; ++v) {
        float bb = lBias[16 * tr + 8 * half + v];
        acc0[v] = bb; acc1[v] = bb;
    }
    #pragma unroll
    for (int k0 = 0; k0 < 64; k0 += 4) {
        int kr = k0 + 2 * half;
        v2f a, b0, b1;
        a.x  = lW[rowA * LDSS + kr];
        a.y  = lW[rowA * LDSS + kr + 1];
        b0.x = lX[kr * LDSS + c0];
        b0.y = lX[(kr + 1) * LDSS + c0];
        b1.x = lX[kr * LDSS + c1];
        b1.y = lX[(kr + 1) * LDSS + c1];
        acc0 = __builtin_amdgcn_wmma_f32_16x16x4_f32(false, a, false, b0, (short)0, acc0, false, false);
        acc1 = __builtin_amdgcn_wmma_f32_16x16x4_f32(false, a, false, b1, (short)0, acc1, false, false);
    }
    #pragma unroll
    for (int v = 0; v < 8; ++v) {
        int r = 16 * tr + 8 * half + v;
        lXp[r * LDSS + c0] = acc0[v];
        lXp[r * LDSS + c1] = acc1[v];
    }
    __syncthreads();   // GEMM1 reads of lW/lX complete -> safe to alias as lS/lA

    // ---- Build S = sum_k D~^-1/2 (A_k + I) D~^-1/2, fused A pass-through --
    const int srow = tid >> 2;           // each thread owns row srow, 16 cols
    const int scol = (tid & 3) << 4;
    #pragma unroll
    for (int j = 0; j < 16; ++j) lS[srow * LDSS + scol + j] = 0.0f;

    for (int k = 0; k < 3; ++k) {
        const size_t aoff = (size_t)((b * 3 + k) * 32 + t) << 12;  // *4096
        const v4f* gA  = (const v4f*)(A + aoff);
        v4f*       gAo = (v4f*)(outA + aoff);
        #pragma unroll
        for (int i = 0; i < 4; ++i) {
            int idx4 = tid + i * 256;
            v4f val = __builtin_nontemporal_load(&gA[idx4]);       // A read once
            __builtin_nontemporal_store(val, &gAo[idx4]);          // fused copy-out
            int n  = idx4 >> 4;
            int m4 = (idx4 & 15) << 2;
            lA[n * LDSS + m4 + 0] = val.x;
            lA[n * LDSS + m4 + 1] = val.y;
            lA[n * LDSS + m4 + 2] = val.z;
            lA[n * LDSS + m4 + 3] = val.w;
        }
        __syncthreads();
        {   // partial row sums: 4 threads per row
            float s = 0.0f;
            #pragma unroll
            for (int j = 0; j < 16; ++j) s += lA[srow * LDSS + scol + j];
            lRed[tid] = s;
        }
        __syncthreads();
        if (tid < 64) {
            float deg = 1.0f + lRed[4 * tid] + lRed[4 * tid + 1]
                             + lRed[4 * tid + 2] + lRed[4 * tid + 3];
            lDinv[tid] = rsqrtf(deg);    // deg >= 1 (A in [0,1), +I) -> safe
        }
        __syncthreads();
        {
            float din = lDinv[srow];
            #pragma unroll
            for (int j = 0; j < 16; ++j) {
                int m = scol + j;
                float av = lA[srow * LDSS + m] + ((m == srow) ? 1.0f : 0.0f);
                lS[srow * LDSS + m] += av * din * lDinv[m];
            }
        }
        __syncthreads();                 // before next k overwrites lA / GEMM2 reads lS
    }

    // ---- GEMM2: out = Xp @ S  (16x16x4 f32 WMMA) --------------------------
    v8f o0, o1;
    #pragma unroll
    for (int v = 0; v < 8; ++v) { o0[v] = 0.0f; o1[v] = 0.0f; }
    #pragma unroll
    for (int k0 = 0; k0 < 64; k0 += 4) {
        int kr = k0 + 2 * half;
        v2f a, b0, b1;
        a.x  = lXp[rowA * LDSS + kr];
        a.y  = lXp[rowA * LDSS + kr + 1];
        b0.x = lS[kr * LDSS + c0];
        b0.y = lS[(kr + 1) * LDSS + c0];
        b1.x = lS[kr * LDSS + c1];
        b1.y = lS[(kr + 1) * LDSS + c1];
        o0 = __builtin_amdgcn_wmma_f32_16x16x4_f32(false, a, false, b0, (short)0, o0, false, false);
        o1 = __builtin_amdgcn_wmma_f32_16x16x4_f32(false, a, false, b1, (short)0, o1, false, false);
    }

    // ---- Store x_out[b, c, t, m] ------------------------------------------
    #pragma unroll
    for (int v = 0; v < 8; ++v) {
        int cc = 16 * tr + 8 * half + v;
        size_t base = (size_t)((b * 64 + cc) * 32 + t) << 6;
        outX[base + c0] = o0[v];
        outX[base + c1] = o1[v];
    }
}

extern "C" void kernel_launch(void* const* d_in, const int* in_sizes, int n_in,
                              void* d_out, int out_size, void* d_ws, size_t ws_size,
                              hipStream_t stream) {
    const float* x    = (const float*)d_in[0];   // (64,64,32,64)
    const float* A    = (const float*)d_in[1];   // (64,3,32,64,64)
    const float* W    = (const float*)d_in[2];   // (64,64)
    const float* bias = (const float*)d_in[3];   // (64,)
    float* outX = (float*)d_out;                         // 64*64*32*64 floats
    float* outA = outX + (size_t)64 * 64 * 32 * 64;      // A pass-through

    dim3 grid(64 * 32);   // one workgroup per (b, t)
    stgcn_fused_kernel<<<grid, 256, 0, stream>>>(x, A, W, bias, outX, outA);
}